// tdeLayerOne_87351044866353
// MI455X (gfx1250) — compile-verified
//
#include <hip/hip_runtime.h>

#ifndef __has_builtin
#define __has_builtin(x) 0
#endif

// Problem constants (from the reference): out[j,k] = in[j + 4k], 32 cols.
#define TDE_DIM      32
#define TDE_DELAY    4
#define TDE_TAIL     ((TDE_DIM - 1) * TDE_DELAY)   // 124

#define BLOCK        256
#define ROWS_PER_BLK 4096
#define TILE_WORDS   (ROWS_PER_BLK + 128)          // 4220 needed, pad to 4224
#define TILE_VEC4    (TILE_WORDS / 4)              // 1056 = 4*256 + 32
#define OUT_V4       (ROWS_PER_BLK * (TDE_DIM / 4))// 32768 float4 per block

typedef __attribute__((ext_vector_type(4))) float v4f;
typedef __attribute__((ext_vector_type(4))) int   v4i;

// Pointer types matching the async-load builtin's signature:
//   (v4i addrspace(1)*, v4i addrspace(3)*, imm offset, imm cpol)
typedef __attribute__((address_space(1))) v4i* g_v4i_ptr;
typedef __attribute__((address_space(3))) v4i* l_v4i_ptr;

__device__ __forceinline__ void async_copy16(const float* g, float* l) {
#if __has_builtin(__builtin_amdgcn_global_load_async_to_lds_b128)
    __builtin_amdgcn_global_load_async_to_lds_b128(
        (g_v4i_ptr)g, (l_v4i_ptr)l, /*offset=*/0, /*cpol=*/0);
#else
    *(v4f*)l = *(const v4f*)g;                      // load_b128 + ds_store_b128
#endif
}

__device__ __forceinline__ void wait_asynccnt0() {
#if __has_builtin(__builtin_amdgcn_s_wait_asynccnt)
    __builtin_amdgcn_s_wait_asynccnt(0);
#else
    asm volatile("s_wait_asynccnt 0" ::: "memory");
#endif
}

__global__ __launch_bounds__(BLOCK)
void tde_gather_kernel(const float* __restrict__ in,
                       float* __restrict__ out,
                       unsigned n, unsigned num_pts) {
    __shared__ float tile[TILE_WORDS];

    const unsigned tid = threadIdx.x;
    const unsigned J0  = blockIdx.x * (unsigned)ROWS_PER_BLK;

    const float* gsrc = in + J0;
    // Interior blocks (1023 of 1024): whole tile in-range, all rows valid.
    const bool interior = (J0 + TILE_WORDS <= n) && (J0 + ROWS_PER_BLK <= num_pts);

    // ---- Stage input window [J0, J0 + TILE_WORDS) into LDS ----------------
    if (interior) {
#pragma unroll
        for (unsigned it = 0; it < 4; ++it) {
            const unsigned w = 4u * (tid + it * BLOCK);
            async_copy16(gsrc + w, tile + w);
        }
        if (tid < (TILE_VEC4 - 4 * BLOCK)) {        // remainder: 32 chunks
            const unsigned w = 4u * (tid + 4 * BLOCK);
            async_copy16(gsrc + w, tile + w);
        }
    } else {
        // Tail block: every word actually consumed satisfies (J0 + w) < n,
        // so guarded 16B-granular loads cover all needed data.
        for (unsigned i = tid; i < TILE_VEC4; i += BLOCK) {
            const unsigned w = 4u * i;
            if (J0 + w + 4 <= n) async_copy16(gsrc + w, tile + w);
        }
    }
    wait_asynccnt0();        // my async loads landed in LDS
    __syncthreads();         // everyone's did (each wave waited before signaling)

    // ---- Emit output: one 128-bit coalesced NT store per thread-iteration --
    // u -> (j_local = u>>3, k0 = u&7); output word offset is exactly 4*u, so a
    // wave's 32 lanes cover 128 consecutive words (512B, fully coalesced) and
    // each thread's store address advances by a constant 4KB per iteration
    // (foldable into global_store immediate offsets). LDS gather base advances
    // by 128B per iteration (foldable into ds offsets).
    float* outp = out + (size_t)J0 * TDE_DIM;
    if (interior) {
#pragma unroll 8
        for (unsigned u = tid; u < OUT_V4; u += BLOCK) {
            const unsigned j_local = u >> 3;
            const unsigned k0      = u & 7;
            const unsigned base    = j_local + 16u * k0;   // in[j + 4*(4*k0+c)]
            v4f v;
            v.x = tile[base];
            v.y = tile[base + 4];
            v.z = tile[base + 8];
            v.w = tile[base + 12];
            // Output is write-once / streaming: non-temporal so the 537MB of
            // stores don't evict the (L2-resident, reused 32x) input.
            __builtin_nontemporal_store(v, (v4f*)(outp + 4u * u));
        }
    } else {
        const unsigned rows = num_pts - J0;                // valid rows in tail
        for (unsigned u = tid; u < OUT_V4; u += BLOCK) {
            const unsigned j_local = u >> 3;
            const unsigned k0      = u & 7;
            if (j_local < rows) {
                const unsigned base = j_local + 16u * k0;
                v4f v;
                v.x = tile[base];
                v.y = tile[base + 4];
                v.z = tile[base + 8];
                v.w = tile[base + 12];
                __builtin_nontemporal_store(v, (v4f*)(outp + 4u * u));
            }
        }
    }
}

extern "C" void kernel_launch(void* const* d_in, const int* in_sizes, int n_in,
                              void* d_out, int out_size, void* d_ws, size_t ws_size,
                              hipStream_t stream) {
    const float* in  = (const float*)d_in[0];
    float*       out = (float*)d_out;
    const unsigned n       = (unsigned)in_sizes[0];          // 4194304
    const unsigned num_pts = n - TDE_TAIL;                   // 4194180
    const unsigned nblocks = (num_pts + ROWS_PER_BLK - 1) / ROWS_PER_BLK;  // 1024
    tde_gather_kernel<<<nblocks, BLOCK, 0, stream>>>(in, out, n, num_pts);
}